// MSE_DQ_FK_56865366999315
// MI455X (gfx1250) — compile-verified
//
#include <hip/hip_runtime.h>
#include <math.h>

typedef float v2f __attribute__((ext_vector_type(2)));
typedef float v8f __attribute__((ext_vector_type(8)));

#define N_J 22
#define TLEN 512
#define BATCH 256
#define NBT (BATCH * TLEN)          // 131072
#define FK_BLOCKS 512               // NBT / 256
#define VEL_ITEMS (NBT * N_J)       // 2883584
#define VEL_BLOCKS 11264            // VEL_ITEMS / 256  (exact)
#define JP_FLOATS (NBT * N_J * 3)   // 8650752

static constexpr int PAR[N_J] = {-1, 0, 1, 2, 3, 0, 5, 6, 7, 0, 9, 10, 11, 11, 13, 14, 15, 11, 17, 18, 19, 11};
static constexpr bool IS_EE[N_J] = {false,false,false,false,true,
                                    false,false,false,true,
                                    false,false,false,true,
                                    false,false,false,true,
                                    false,false,false,true,false};

__device__ inline float wave_sum(float v) {
#pragma unroll
  for (int o = 16; o > 0; o >>= 1) v += __shfl_xor(v, o, 32);
  return v;
}

// ---------------------------------------------------------------------------
// Kernel 1: per-(b,t) dual-quat -> local rot -> FK chain -> fused SSE losses.
// Writes joint positions to jp_out (for the velocity pass) and 4 partial sums
// per block: [0]=ee_pos, [1]=ee_rot, [2]=reg_pos, [3]=reg_rot.
// ---------------------------------------------------------------------------
__global__ void fk_loss_kernel(const float* __restrict__ ik,
                               const float* __restrict__ tpose,
                               const float* __restrict__ trot,
                               const float* __restrict__ offs,
                               float* __restrict__ jp_out,
                               float* __restrict__ part) {
  const int tid = threadIdx.x;
  const int idx = blockIdx.x * 256 + tid;   // [0, NBT)
  const int b = idx / TLEN;
  const int t = idx - b * TLEN;

  // ---- load + normalize real quaternion part only (dual part is unused) ----
  float q[N_J][4];
  const size_t ikbase = ((size_t)b * (N_J * 8)) * TLEN + (size_t)t;
#pragma unroll
  for (int j = 0; j < N_J; ++j) {
    const float w = ik[ikbase + (size_t)(j * 8 + 0) * TLEN];
    const float x = ik[ikbase + (size_t)(j * 8 + 1) * TLEN];
    const float y = ik[ikbase + (size_t)(j * 8 + 2) * TLEN];
    const float z = ik[ikbase + (size_t)(j * 8 + 3) * TLEN];
    const float n = sqrtf(w * w + x * x + y * y + z * z) + 1e-8f;
    const float inv = 1.0f / n;
    q[j][0] = w * inv; q[j][1] = x * inv; q[j][2] = y * inv; q[j][3] = z * inv;
  }

  const size_t tbase = (size_t)idx * N_J;   // (b*T + t) * 22
  __builtin_prefetch(trot + tbase * 9, 0, 0);       // global_prefetch_b8
  __builtin_prefetch(trot + tbase * 9 + 128, 0, 0);

  float aEEp = 0.f, aEEr = 0.f, aRGp = 0.f, aRGr = 0.f;
  float gR[N_J][9];
  float gP[N_J][3];

#pragma unroll
  for (int j = 0; j < N_J; ++j) {
    // local rotation: j==0 -> q[0]; else conj(q[parent]) * q[j]
    float lw, lx, ly, lz;
    if (j == 0) {
      lw = q[0][0]; lx = q[0][1]; ly = q[0][2]; lz = q[0][3];
    } else {
      const int p = PAR[j];
      const float aw = q[p][0], ax = -q[p][1], ay = -q[p][2], az = -q[p][3];
      const float bw = q[j][0], bx = q[j][1], by = q[j][2], bz = q[j][3];
      lw = aw * bw - ax * bx - ay * by - az * bz;
      lx = aw * bx + ax * bw + ay * bz - az * by;
      ly = aw * by - ax * bz + ay * bw + az * bx;
      lz = aw * bz + ax * by - ay * bx + az * bw;
    }
    // quat -> 3x3 (row major)
    float m[9];
    m[0] = 1.f - 2.f * (ly * ly + lz * lz);
    m[1] = 2.f * (lx * ly - lw * lz);
    m[2] = 2.f * (lx * lz + lw * ly);
    m[3] = 2.f * (lx * ly + lw * lz);
    m[4] = 1.f - 2.f * (lx * lx + lz * lz);
    m[5] = 2.f * (ly * lz - lw * lx);
    m[6] = 2.f * (lx * lz - lw * ly);
    m[7] = 2.f * (ly * lz + lw * lx);
    m[8] = 1.f - 2.f * (lx * lx + ly * ly);

    if (j == 0) {
#pragma unroll
      for (int k = 0; k < 9; ++k) gR[0][k] = m[k];
      const float* rp = tpose + tbase * 3;   // joint 0 position = target root
      gP[0][0] = rp[0]; gP[0][1] = rp[1]; gP[0][2] = rp[2];
    } else {
      const int p = PAR[j];
#pragma unroll
      for (int r = 0; r < 3; ++r)
#pragma unroll
        for (int c = 0; c < 3; ++c)
          gR[j][r * 3 + c] = gR[p][r * 3 + 0] * m[0 * 3 + c] +
                             gR[p][r * 3 + 1] * m[1 * 3 + c] +
                             gR[p][r * 3 + 2] * m[2 * 3 + c];
      const float ox = offs[j * 3 + 0], oy = offs[j * 3 + 1], oz = offs[j * 3 + 2];
#pragma unroll
      for (int r = 0; r < 3; ++r)
        gP[j][r] = gP[p][r] + gR[p][r * 3 + 0] * ox + gR[p][r * 3 + 1] * oy +
                   gR[p][r * 3 + 2] * oz;
    }

    // stash joint position for the velocity pass
    float* o = jp_out + (tbase + (size_t)j) * 3;
    o[0] = gP[j][0]; o[1] = gP[j][1]; o[2] = gP[j][2];

    // fused SSE accumulation (joint 0 appears in neither loss term)
    if (j != 0) {
      const float* tp = tpose + (tbase + (size_t)j) * 3;
      const float* tr = trot + (tbase + (size_t)j) * 9;
      const float d0 = gP[j][0] - tp[0];
      const float d1 = gP[j][1] - tp[1];
      const float d2 = gP[j][2] - tp[2];
      const float sp = d0 * d0 + d1 * d1 + d2 * d2;
      float sr = 0.f;
#pragma unroll
      for (int k = 0; k < 9; ++k) {
        const float dr = gR[j][k] - tr[k];
        sr += dr * dr;
      }
      if (IS_EE[j]) { aEEp += sp; aEEr += sr; }
      else          { aRGp += sp; aRGr += sr; }
    }
  }

  // ---- deterministic block reduction: wave shuffles + LDS combine ----
  __shared__ float red[8][4];
  aEEp = wave_sum(aEEp); aEEr = wave_sum(aEEr);
  aRGp = wave_sum(aRGp); aRGr = wave_sum(aRGr);
  const int lane = tid & 31, wv = tid >> 5;
  if (lane == 0) {
    red[wv][0] = aEEp; red[wv][1] = aEEr; red[wv][2] = aRGp; red[wv][3] = aRGr;
  }
  __syncthreads();
  if (tid == 0) {
    float s0 = 0.f, s1 = 0.f, s2 = 0.f, s3 = 0.f;
#pragma unroll
    for (int w = 0; w < 8; ++w) {
      s0 += red[w][0]; s1 += red[w][1]; s2 += red[w][2]; s3 += red[w][3];
    }
    part[0 * FK_BLOCKS + blockIdx.x] = s0;
    part[1 * FK_BLOCKS + blockIdx.x] = s1;
    part[2 * FK_BLOCKS + blockIdx.x] = s2;
    part[3 * FK_BLOCKS + blockIdx.x] = s3;
  }
}

// ---------------------------------------------------------------------------
// Kernel 2: velocity losses. One thread per (b,t,j).
// e_d = r(t+d) - r(t) where r = target_pos - fk_pos (contiguous 12B loads).
// ---------------------------------------------------------------------------
__global__ void vel_kernel(const float* __restrict__ tpose,
                           const float* __restrict__ jp,
                           float* __restrict__ part) {
  const int tid = threadIdx.x;
  const size_t idx = (size_t)blockIdx.x * 256 + tid;  // [0, VEL_ITEMS)
  const int t = (int)((idx / N_J) % TLEN);

  const float* tp = tpose + idx * 3;
  const float* jq = jp + idx * 3;
  __builtin_prefetch(tp + 5 * N_J * 3, 0, 0);
  const float rx = tp[0] - jq[0];
  const float ry = tp[1] - jq[1];
  const float rz = tp[2] - jq[2];

  float acc[3] = {0.f, 0.f, 0.f};
  const int ds[3] = {1, 3, 5};
#pragma unroll
  for (int i = 0; i < 3; ++i) {
    const int d = ds[i];
    if (t + d < TLEN) {
      const int off = d * N_J * 3;   // advance t by d
      const float ex = (tp[off + 0] - jq[off + 0]) - rx;
      const float ey = (tp[off + 1] - jq[off + 1]) - ry;
      const float ez = (tp[off + 2] - jq[off + 2]) - rz;
      acc[i] = sqrtf(ex * ex + ey * ey + ez * ez);
    }
  }

  __shared__ float red[8][3];
  const int lane = tid & 31, wv = tid >> 5;
  float v0 = wave_sum(acc[0]), v1 = wave_sum(acc[1]), v2 = wave_sum(acc[2]);
  if (lane == 0) { red[wv][0] = v0; red[wv][1] = v1; red[wv][2] = v2; }
  __syncthreads();
  if (tid == 0) {
    float s0 = 0.f, s1 = 0.f, s2 = 0.f;
#pragma unroll
    for (int w = 0; w < 8; ++w) { s0 += red[w][0]; s1 += red[w][1]; s2 += red[w][2]; }
    part[0 * VEL_BLOCKS + blockIdx.x] = s0;
    part[1 * VEL_BLOCKS + blockIdx.x] = s1;
    part[2 * VEL_BLOCKS + blockIdx.x] = s2;
  }
}

// ---------------------------------------------------------------------------
// Kernel 3: single-wave final reduction on the matrix unit.
// Chained V_WMMA_F32_16X16X4_F32 with all-ones B: each op folds 64 partials
// (2 per lane in the 16x4 A tile) into the 16 row-sums of the f32 accumulator.
// Fixed operation order -> bit-deterministic across replays.
// ---------------------------------------------------------------------------
__device__ inline float wmma_sum_chunks(const float* __restrict__ p,
                                        int n_chunks, int lane) {
  v8f acc = {};
  v2f ones; ones.x = 1.0f; ones.y = 1.0f;
  for (int it = 0; it < n_chunks; ++it) {
    v2f a;
    a.x = p[it * 64 + lane * 2 + 0];
    a.y = p[it * 64 + lane * 2 + 1];
    // D = A(16x4) * ones(4x16) + C : every column of D holds the row-sums.
    acc = __builtin_amdgcn_wmma_f32_16x16x4_f32(
        /*neg_a=*/false, a, /*neg_b=*/false, ones,
        /*c_mod=*/(short)0, acc, /*reuse_a=*/false, /*reuse_b=*/false);
  }
  float s = acc[0] + acc[1] + acc[2] + acc[3] + acc[4] + acc[5] + acc[6] + acc[7];
  // lanes 0-15 hold rows M=0..7 summed, lanes 16-31 hold rows M=8..15 summed
  s += __shfl_xor(s, 16, 32);
  return s;
}

__global__ void final_reduce_kernel(const float* __restrict__ fkp,
                                    const float* __restrict__ velp,
                                    float* __restrict__ out) {
  const int lane = threadIdx.x;  // exactly one wave32, EXEC all ones

  float s[7];
#pragma unroll
  for (int c = 0; c < 4; ++c)
    s[c] = wmma_sum_chunks(fkp + c * FK_BLOCKS, FK_BLOCKS / 64, lane);
#pragma unroll
  for (int c = 0; c < 3; ++c)
    s[4 + c] = wmma_sum_chunks(velp + c * VEL_BLOCKS, VEL_BLOCKS / 64, lane);

  if (lane == 0) {
    const float nbt = (float)NBT;
    const float cEEp = nbt * 5.f * 3.f;
    const float cEEr = nbt * 5.f * 9.f;
    const float cRGp = nbt * 16.f * 3.f;
    const float cRGr = nbt * 16.f * 9.f;
    const float cV1 = (float)BATCH * (float)(TLEN - 1) * (float)N_J;
    const float cV3 = (float)BATCH * (float)(TLEN - 3) * (float)N_J;
    const float cV5 = (float)BATCH * (float)(TLEN - 5) * (float)N_J;
    const float loss_ee = s[0] / cEEp + s[1] / cEEr;
    const float loss_reg = s[2] / cRGp + s[3] / cRGr;
    const float loss_vel = s[4] / cV1 + s[5] / cV3 + s[6] / cV5;
    out[0] = 10.0f * loss_ee + 1.0f * loss_reg + 10.0f * loss_vel;
  }
}

// ---------------------------------------------------------------------------
extern "C" void kernel_launch(void* const* d_in, const int* in_sizes, int n_in,
                              void* d_out, int out_size, void* d_ws, size_t ws_size,
                              hipStream_t stream) {
  const float* ik    = (const float*)d_in[0];  // (B, 176, T)
  const float* tpose = (const float*)d_in[1];  // (B, T, 22, 3)
  const float* trot  = (const float*)d_in[2];  // (B, T, 22, 3, 3)
  const float* offs  = (const float*)d_in[3];  // (22, 3)

  float* ws = (float*)d_ws;
  float* jp   = ws;                         // JP_FLOATS floats (~34.6 MB)
  float* fkp  = ws + JP_FLOATS;             // 4 * FK_BLOCKS floats
  float* velp = fkp + 4 * FK_BLOCKS;        // 3 * VEL_BLOCKS floats

  fk_loss_kernel<<<FK_BLOCKS, 256, 0, stream>>>(ik, tpose, trot, offs, jp, fkp);
  vel_kernel<<<VEL_BLOCKS, 256, 0, stream>>>(tpose, jp, velp);
  final_reduce_kernel<<<1, 32, 0, stream>>>(fkp, velp, (float*)d_out);
}